// BucketAdjustedHinge_62878321213999
// MI455X (gfx1250) — compile-verified
//
#include <hip/hip_runtime.h>

#define KB 20   // base knot count
#define KA 20   // per-bucket knot count
#define NB 64   // buckets
#define SK 22   // padded row stride (even -> every [k,k+1] ds_load_b64 stays 8B aligned)

#if defined(__gfx1250__) && __has_builtin(__builtin_amdgcn_global_load_async_to_lds_b32)
#define USE_ASYNC_LDS 1
#else
#define USE_ASYNC_LDS 0
#endif

typedef float vfloat4 __attribute__((ext_vector_type(4)));
typedef int   vint4   __attribute__((ext_vector_type(4)));
typedef float vfloat2 __attribute__((ext_vector_type(2)));

__device__ __forceinline__ void stage_b32(const float* __restrict__ gsrc,
                                          float* __restrict__ lds_dst) {
#if USE_ASYNC_LDS
    __builtin_amdgcn_global_load_async_to_lds_b32(
        (__attribute__((address_space(1))) int*)gsrc,
        (__attribute__((address_space(3))) int*)lds_dst,
        /*offset=*/0, /*cpol=*/0);
#else
    *lds_dst = *gsrc;
#endif
}

__global__ __launch_bounds__(256)
void BucketAdjustedHinge_kernel(const float* __restrict__ x,
                                const int*   __restrict__ bidx,
                                const float* __restrict__ base_knots,
                                const float* __restrict__ base_w,
                                const float* __restrict__ base_b,
                                const float* __restrict__ adj_knots,
                                const float* __restrict__ adj_w,
                                const float* __restrict__ adj_b,
                                float*       __restrict__ out,
                                int n) {
    // knots and weights in separate contiguous rows so ds_load_b64 yields a
    // packed <2 x float> of two consecutive knots (v_pk_* operand, no shuffle)
    __shared__ float s_ak[NB * SK];
    __shared__ float s_aw[NB * SK];
    __shared__ float s_adjb[NB];

    const int tid = threadIdx.x;

    // ---- stage bucket tables into LDS via async copy (ASYNCcnt path) ----
    for (int t = tid; t < NB * KA; t += blockDim.x) {
        const int b = t / KA;
        const int k = t - b * KA;
        stage_b32(adj_knots + t, &s_ak[b * SK + k]);
        stage_b32(adj_w     + t, &s_aw[b * SK + k]);
    }
    for (int t = tid; t < NB; t += blockDim.x) {
        stage_b32(adj_b + t, &s_adjb[t]);
    }
#if USE_ASYNC_LDS
    asm volatile("s_wait_asynccnt 0" ::: "memory");
#endif
    __syncthreads();

    // ---- base table: uniform -> scalar loads, packed pairs in SGPRs ----
    vfloat2 bk2[KB / 2], bw2[KB / 2];
    const vfloat2* gk2 = (const vfloat2*)base_knots;
    const vfloat2* gw2 = (const vfloat2*)base_w;
#pragma unroll
    for (int j = 0; j < KB / 2; ++j) { bk2[j] = gk2[j]; bw2[j] = gw2[j]; }
    const float bb = base_b[0];
    const vfloat2 z2 = {0.0f, 0.0f};

    const int gid    = blockIdx.x * blockDim.x + tid;
    const int stride = gridDim.x * blockDim.x;
    const int nvec   = n >> 2;

    const vfloat4* __restrict__ x4 = (const vfloat4*)x;
    const vint4*   __restrict__ i4 = (const vint4*)bidx;
    vfloat4*       __restrict__ o4 = (vfloat4*)out;

    for (int i = gid; i < nvec; i += stride) {
        const vfloat4 xv = __builtin_nontemporal_load(&x4[i]);
        const vint4   bv = __builtin_nontemporal_load(&i4[i]);

        vfloat4 r;
#pragma unroll
        for (int e = 0; e < 4; ++e) {
            const float xe = xv[e];
            const int   be = bv[e];
            const vfloat2* __restrict__ rowk = (const vfloat2*)&s_ak[be * SK];
            const vfloat2* __restrict__ roww = (const vfloat2*)&s_aw[be * SK];

            vfloat2 acc = {bb + s_adjb[be], 0.0f};  // 2-wide accumulator
#pragma unroll
            for (int j = 0; j < KA / 2; ++j) {      // 2 knots per packed op
                const vfloat2 kk = rowk[j];         // ds_load_b64
                const vfloat2 ww = roww[j];         // ds_load_b64
                vfloat2 t = xe - kk;                // v_pk_add_f32 (neg, splat x)
                t = __builtin_elementwise_max(t, z2);
                acc = __builtin_elementwise_fma(t, ww, acc);  // v_pk_fma_f32
            }
#pragma unroll
            for (int j = 0; j < KB / 2; ++j) {      // SGPR-pair base table
                vfloat2 t = xe - bk2[j];
                t = __builtin_elementwise_max(t, z2);
                acc = __builtin_elementwise_fma(t, bw2[j], acc);
            }
            r[e] = acc[0] + acc[1];                 // horizontal reduce
        }

        __builtin_nontemporal_store(r, &o4[i]);
    }

    // scalar tail (n not divisible by 4)
    for (int i = nvec * 4 + gid; i < n; i += stride) {
        const float xe = x[i];
        const int   b  = bidx[i];
        float a = bb + s_adjb[b];
#pragma unroll
        for (int k = 0; k < KA; ++k)
            a = fmaf(fmaxf(xe - s_ak[b * SK + k], 0.0f), s_aw[b * SK + k], a);
        const float* bkf = (const float*)bk2;
        const float* bwf = (const float*)bw2;
#pragma unroll
        for (int k = 0; k < KB; ++k)
            a = fmaf(fmaxf(xe - bkf[k], 0.0f), bwf[k], a);
        out[i] = a;
    }
}

extern "C" void kernel_launch(void* const* d_in, const int* in_sizes, int n_in,
                              void* d_out, int out_size, void* d_ws, size_t ws_size,
                              hipStream_t stream) {
    const float* x          = (const float*)d_in[0];
    const int*   bucket_idx = (const int*)  d_in[1];
    const float* base_knots = (const float*)d_in[2];
    const float* base_w     = (const float*)d_in[3];
    const float* base_b     = (const float*)d_in[4];
    const float* adj_knots  = (const float*)d_in[5];
    const float* adj_w      = (const float*)d_in[6];
    const float* adj_b      = (const float*)d_in[7];
    float* out = (float*)d_out;

    const int n       = in_sizes[0];
    const int threads = 256;                 // 8 wave32 waves per block
    int blocks = (n / 4 + threads - 1) / threads;
    if (blocks > 2048) blocks = 2048;        // grid-stride; amortize LDS staging
    if (blocks < 1) blocks = 1;

    BucketAdjustedHinge_kernel<<<blocks, threads, 0, stream>>>(
        x, bucket_idx, base_knots, base_w, base_b,
        adj_knots, adj_w, adj_b, out, n);
}